// SonataTransformerBlock_55722905698813
// MI455X (gfx1250) — compile-verified
//
#include <hip/hip_runtime.h>
#include <math.h>

// SonataTransformerBlock for MI455X (gfx1250, wave32).
// fp32 end-to-end; all GEMMs use V_WMMA_F32_16X16X4_F32.
// k_proj stages its LDS tile with the Tensor Data Mover (TENSOR_LOAD_TO_LDS).
// N (=100000) must be a multiple of 16 (true for this harness).

#define DIM   512
#define GRP   4
#define GD    128
#define KNB   16
#define HID   1024
#define TILE  16
#define LDPAD 4
#define LDA   (DIM + LDPAD)   // 516 floats: row stride 2064B -> 4-bank skew per row
#define LDH   (HID + LDPAD)   // 1028 floats

typedef float v2f __attribute__((ext_vector_type(2)));
typedef float v8f __attribute__((ext_vector_type(8)));
typedef unsigned int u32x4 __attribute__((ext_vector_type(4)));
typedef int i32x4 __attribute__((ext_vector_type(4)));
typedef int i32x8 __attribute__((ext_vector_type(8)));

__device__ __forceinline__ v8f wmma4(v2f a, v2f b, v8f c) {
  // D(16x16,f32) = A(16x4,f32) * B(4x16,f32) + C
  return __builtin_amdgcn_wmma_f32_16x16x4_f32(
      /*neg_a=*/false, a, /*neg_b=*/false, b,
      /*c_mod=*/(short)0, c, /*reuse_a=*/false, /*reuse_b=*/false);
}

// One 16x16 output tile, K-loop in steps of 4, accumulating onto `acc`.
// ap = &A_lds[row=ml][k0 + kb]   (kb = 2*(lane>>4): half-wave K split)
// bp = &W[o0+ml][k0 + kb]        (B[k][n] = W[n][k], row-major global)
__device__ __forceinline__ v8f gemm16(const float* ap, const float* bp, int kdim, v8f acc) {
  #pragma unroll 4
  for (int kc = 0; kc < kdim; kc += 4) {
    v2f a = *(const v2f*)(ap + kc);
    v2f b = *(const v2f*)(bp + kc);
    acc = wmma4(a, b, acc);
  }
  return acc;
}

// LayerNorm 16-token tile into padded LDS; each wave handles 2 tokens.
__device__ __forceinline__ void ln_tile_to_lds(const float* __restrict__ src,
                                               const float* __restrict__ gam,
                                               const float* __restrict__ bet,
                                               float* lds, int t0, int lane, int wid) {
  #pragma unroll
  for (int i = 0; i < 2; ++i) {
    const int tt = 2 * wid + i;
    const float4* xr = (const float4*)(src + (size_t)(t0 + tt) * DIM + lane * 16);
    float4 x4[4];
    #pragma unroll
    for (int j = 0; j < 4; ++j) x4[j] = xr[j];
    float s = 0.f, ss = 0.f;
    #pragma unroll
    for (int j = 0; j < 4; ++j) {
      s  += x4[j].x + x4[j].y + x4[j].z + x4[j].w;
      ss += x4[j].x * x4[j].x + x4[j].y * x4[j].y + x4[j].z * x4[j].z + x4[j].w * x4[j].w;
    }
    #pragma unroll
    for (int off = 16; off; off >>= 1) { s += __shfl_xor(s, off); ss += __shfl_xor(ss, off); }
    const float mu   = s * (1.0f / DIM);
    const float rstd = rsqrtf(ss * (1.0f / DIM) - mu * mu + 1e-5f);
    const float4* gp = (const float4*)(gam + lane * 16);
    const float4* bp = (const float4*)(bet + lane * 16);
    float4* dst = (float4*)(lds + tt * LDA + lane * 16);
    #pragma unroll
    for (int j = 0; j < 4; ++j) {
      float4 xv = x4[j], gv = gp[j], bv = bp[j], o;
      o.x = (xv.x - mu) * rstd * gv.x + bv.x;
      o.y = (xv.y - mu) * rstd * gv.y + bv.y;
      o.z = (xv.z - mu) * rstd * gv.z + bv.z;
      o.w = (xv.w - mu) * rstd * gv.w + bv.w;
      dst[j] = o;
    }
  }
}

// ---------------- Pass 1: LN1 + grouped QKV projections ----------------
__global__ void k_ln_qkv(const float* __restrict__ feat,
                         const float* __restrict__ Wq, const float* __restrict__ Wk,
                         const float* __restrict__ Wv,
                         const float* __restrict__ g1, const float* __restrict__ b1,
                         float* __restrict__ qb, float* __restrict__ kb_,
                         float* __restrict__ vb, int n) {
  extern __shared__ float smem[];                     // TILE x LDA
  const int t0 = blockIdx.x * TILE;
  if (t0 >= n) return;
  const int lane = threadIdx.x & 31;
  const int wid  = threadIdx.x >> 5;
  ln_tile_to_lds(feat, g1, b1, smem, t0, lane, wid);
  __syncthreads();

  const int ml = lane & 15;
  const int hl = lane >> 4;
  const int kb = hl * 2;
  const v8f zero = {};
  // 3 matrices x 4 groups x 8 n-tiles = 96 tiles of 16x16 over 8 waves
  for (int ti = wid; ti < 96; ti += 8) {
    const int msel = ti >> 5;
    const int rem  = ti & 31;
    const int gg   = rem >> 3;
    const int nt   = rem & 7;
    const float* W = (msel == 0) ? Wq : (msel == 1) ? Wk : Wv;
    const float* bp = W + ((size_t)gg * GD + nt * 16 + ml) * GD + kb;   // W[g][o][d]
    __builtin_prefetch(bp, 0, 1);
    const float* ap = smem + ml * LDA + gg * GD + kb;
    v8f acc = gemm16(ap, bp, GD, zero);
    float* ob = (msel == 0) ? qb : (msel == 1) ? kb_ : vb;
    #pragma unroll
    for (int r = 0; r < 8; ++r)
      ob[(size_t)(t0 + r + 8 * hl) * DIM + gg * GD + nt * 16 + ml] = acc[r];
  }
}

// ---------------- Pass 2: K=16 neighbor attention (wave per token) ------
__global__ void k_attn(const float* __restrict__ q, const float* __restrict__ kbuf,
                       const float* __restrict__ vbuf, const int* __restrict__ nbr,
                       float* __restrict__ attn, int n) {
  const int lane = threadIdx.x & 31;
  const int wid  = threadIdx.x >> 5;
  const int t = blockIdx.x * 8 + wid;
  if (t >= n) return;
  // lane owns dims [16*lane, 16*lane+16); head g = lanes [8g, 8g+8)
  const float4* qp = (const float4*)(q + (size_t)t * DIM + lane * 16);
  float4 q4[4];
  #pragma unroll
  for (int j = 0; j < 4; ++j) q4[j] = qp[j];

  int   nb[KNB];
  float sc[KNB];
  #pragma unroll
  for (int k = 0; k < KNB; ++k) nb[k] = nbr[t * KNB + k];

  #pragma unroll
  for (int k = 0; k < KNB; ++k) {
    const float4* kp = (const float4*)(kbuf + (size_t)nb[k] * DIM + lane * 16);
    float p = 0.f;
    #pragma unroll
    for (int j = 0; j < 4; ++j) {
      float4 kv = kp[j];
      p += q4[j].x * kv.x + q4[j].y * kv.y + q4[j].z * kv.z + q4[j].w * kv.w;
    }
    p += __shfl_xor(p, 1);  // reduce within 8-lane head cluster
    p += __shfl_xor(p, 2);
    p += __shfl_xor(p, 4);
    sc[k] = p * 0.08838834764831845f;  // 1/sqrt(128)
  }
  float m = sc[0];
  #pragma unroll
  for (int k = 1; k < KNB; ++k) m = fmaxf(m, sc[k]);
  float den = 0.f;
  #pragma unroll
  for (int k = 0; k < KNB; ++k) { sc[k] = expf(sc[k] - m); den += sc[k]; }
  const float inv = 1.0f / den;

  float4 acc[4];
  #pragma unroll
  for (int j = 0; j < 4; ++j) acc[j] = make_float4(0.f, 0.f, 0.f, 0.f);
  #pragma unroll
  for (int k = 0; k < KNB; ++k) {
    const float w = sc[k] * inv;
    const float4* vp = (const float4*)(vbuf + (size_t)nb[k] * DIM + lane * 16);
    #pragma unroll
    for (int j = 0; j < 4; ++j) {
      float4 vv = vp[j];
      acc[j].x += w * vv.x; acc[j].y += w * vv.y;
      acc[j].z += w * vv.z; acc[j].w += w * vv.w;
    }
  }
  float4* op = (float4*)(attn + (size_t)t * DIM + lane * 16);
  #pragma unroll
  for (int j = 0; j < 4; ++j) op[j] = acc[j];
}

// ---------------- Pass 3: out = attn @ Wo^T + bo + residual -------------
// The 16x512 attn tile is staged into LDS by the Tensor Data Mover with
// LDS padding (pad_interval=256 DWORDs, pad_amount=2 DWORDs), producing a
// 516-float row stride with a 2-float gap after column 256:
//   lds_off(row, col) = row*516 + col + 2*(col >= 256)
__global__ void k_proj(const float* __restrict__ attn, const float* __restrict__ Wo,
                       const float* __restrict__ bo, const float* __restrict__ feat,
                       float* __restrict__ out, int n) {
  extern __shared__ float smem[];                     // TILE x LDA (TDM-padded)
  const int t0 = blockIdx.x * TILE;
  if (t0 >= n) return;
  const int lane = threadIdx.x & 31;
  const int wid  = threadIdx.x >> 5;

  if (wid == 0) {
    // D# group 0: count=1 | lds_addr=0 (dynamic LDS base) | global_addr | type=2
    const unsigned long long ga = (unsigned long long)(const void*)(attn + (size_t)t0 * DIM);
    u32x4 g0;
    g0[0] = 1u;                                         // count=1, user descriptor
    g0[1] = 0u;                                         // lds_addr (bytes)
    g0[2] = (unsigned)(ga & 0xffffffffu);               // global_addr[31:0]
    g0[3] = (unsigned)((ga >> 32) & 0x01ffffffu) | (2u << 30);  // addr[56:32] | type=2
    // D# group 1: data_size=4B, pad_enable, pad_interval=256 DW, pad_amount=2 DW
    const unsigned flags = (2u << 16) | (1u << 20) | (7u << 22) | (1u << 25);
    i32x8 g1;
    g1[0] = (int)flags;
    g1[1] = (int)((512u & 0xffffu) << 16);              // abar_addr=0 | tensor_dim0 lo16
    g1[2] = (int)(((unsigned)n & 0xffffu) << 16);       // tensor_dim0 hi16(=0) | dim1 lo16
    g1[3] = (int)(((unsigned)n >> 16) | (512u << 16));  // dim1 hi16 | tile_dim0=512
    g1[4] = (int)16u;                                   // tile_dim1=16, tile_dim2=0
    g1[5] = (int)512u;                                  // tensor_dim0_stride lo32 = 512
    g1[6] = 0;                                          // stride hi16 | dim1_stride lo16
    g1[7] = 0;
    const i32x4 gz4 = {0, 0, 0, 0};
    const i32x8 gz8 = {0, 0, 0, 0, 0, 0, 0, 0};
    __builtin_amdgcn_tensor_load_to_lds(g0, g1, gz4, gz4, gz8, 0);
    __builtin_amdgcn_s_wait_tensorcnt(0);
  }
  __syncthreads();

  const int ml = lane & 15;
  const int hl = lane >> 4;
  const int kb = hl * 2;
  const v8f zero = {};
  for (int nt = wid; nt < 32; nt += 8) {
    const int o = nt * 16 + ml;
    const float* bp = Wo + (size_t)o * DIM + kb;        // Wo[o][d]
    __builtin_prefetch(bp, 0, 1);
    const float* ap = smem + ml * LDA + kb;
    v8f acc = gemm16(ap,       bp,       256, zero);    // cols [0,256)
    acc     = gemm16(ap + 258, bp + 256, 256, acc);     // cols [256,512), +2 pad skip
    const float bias = bo[o];
    #pragma unroll
    for (int r = 0; r < 8; ++r) {
      const int tok = t0 + r + 8 * hl;
      out[(size_t)tok * DIM + o] = acc[r] + bias + feat[(size_t)tok * DIM + o];
    }
  }
}

// ---------------- Pass 4: LN2 + MLP (GELU) + residual -------------------
__global__ void k_mlp(const float* __restrict__ outb,
                      const float* __restrict__ g2, const float* __restrict__ b2g,
                      const float* __restrict__ W1, const float* __restrict__ b1v,
                      const float* __restrict__ W2, const float* __restrict__ b2v,
                      float* __restrict__ y, int n) {
  extern __shared__ float smem[];
  float* xln = smem;                 // TILE x LDA
  float* hb  = smem + TILE * LDA;    // TILE x LDH
  const int t0 = blockIdx.x * TILE;
  if (t0 >= n) return;
  const int lane = threadIdx.x & 31;
  const int wid  = threadIdx.x >> 5;
  ln_tile_to_lds(outb, g2, b2g, xln, t0, lane, wid);
  __syncthreads();

  const int ml = lane & 15;
  const int hl = lane >> 4;
  const int kb = hl * 2;
  const v8f zero = {};
  // h = GELU(xln @ W1^T + b1): 64 tiles of 16x16, K = 512
  for (int nt = wid; nt < 64; nt += 8) {
    const int o = nt * 16 + ml;
    const float* bp = W1 + (size_t)o * DIM + kb;
    __builtin_prefetch(bp, 0, 1);
    v8f acc = gemm16(xln + ml * LDA + kb, bp, DIM, zero);
    const float bias = b1v[o];
    #pragma unroll
    for (int r = 0; r < 8; ++r) {
      const float t = acc[r] + bias;
      hb[(r + 8 * hl) * LDH + o] = 0.5f * t * (1.0f + erff(t * 0.70710678118654752f));
    }
  }
  __syncthreads();
  // y = h @ W2^T + b2 + out: 32 tiles of 16x16, K = 1024
  for (int nt = wid; nt < 32; nt += 8) {
    const int o = nt * 16 + ml;
    const float* bp = W2 + (size_t)o * HID + kb;
    __builtin_prefetch(bp, 0, 1);
    v8f acc = gemm16(hb + ml * LDH + kb, bp, HID, zero);
    const float bias = b2v[o];
    #pragma unroll
    for (int r = 0; r < 8; ++r) {
      const int tok = t0 + r + 8 * hl;
      y[(size_t)tok * DIM + o] = acc[r] + bias + outb[(size_t)tok * DIM + o];
    }
  }
}

extern "C" void kernel_launch(void* const* d_in, const int* in_sizes, int n_in,
                              void* d_out, int out_size, void* d_ws, size_t ws_size,
                              hipStream_t stream) {
  const float* feat = (const float*)d_in[0];
  // d_in[1] (coords) unused by the reference
  const int*   nbrs = (const int*)d_in[2];
  const float* Wq   = (const float*)d_in[3];
  const float* Wk   = (const float*)d_in[4];
  const float* Wv   = (const float*)d_in[5];
  const float* Wo   = (const float*)d_in[6];
  const float* bo   = (const float*)d_in[7];
  const float* ln1g = (const float*)d_in[8];
  const float* ln1b = (const float*)d_in[9];
  const float* ln2g = (const float*)d_in[10];
  const float* ln2b = (const float*)d_in[11];
  const float* W1   = (const float*)d_in[12];
  const float* b1   = (const float*)d_in[13];
  const float* W2   = (const float*)d_in[14];
  const float* b2   = (const float*)d_in[15];
  float* y = (float*)d_out;

  const int n = in_sizes[0] / DIM;

  // Workspace: q | k | v, each N*DIM fp32. attn aliases q (q is per-token
  // consumed before attn[t] is written); out aliases k (dead after k_attn).
  float* qb   = (float*)d_ws;
  float* kb   = qb + (size_t)n * DIM;
  float* vb   = kb + (size_t)n * DIM;
  float* attn = qb;
  float* outb = kb;

  const int ntile = (n + TILE - 1) / TILE;
  const size_t sh1 = (size_t)TILE * LDA * sizeof(float);
  const size_t sh3 = (size_t)TILE * (LDA + LDH) * sizeof(float);

  k_ln_qkv<<<ntile, 256, sh1, stream>>>(feat, Wq, Wk, Wv, ln1g, ln1b, qb, kb, vb, n);
  k_attn <<<(n + 7) / 8, 256, 0, stream>>>(qb, kb, vb, nbrs, attn, n);
  k_proj <<<ntile, 256, sh1, stream>>>(attn, Wo, bo, feat, outb, n);
  k_mlp  <<<ntile, 256, sh3, stream>>>(outb, ln2g, ln2b, W1, b1, W2, b2, y, n);
}